// VectorQuantizer_90967407329783
// MI455X (gfx1250) — compile-verified
//
#include <hip/hip_runtime.h>
#include <hip/hip_bf16.h>
#include <cmath>

typedef __attribute__((ext_vector_type(2))) float v2f;
typedef __attribute__((ext_vector_type(8))) float v8f;

#define NUM_K   512
#define DIM     64
#define HW      4096            // H*W
#define PLANE   (64*4096)       // C*H*W elements per batch image
#define WROW    68              // padded LDS row stride (bank-conflict free)
#define OUT_OFF 1
#define PERP_OFF 8388609
#define ENC_OFF 8388610
#define NELEM   8388608.0f      // N*D
#define SMEM_FLOATS (NUM_K*WROW + NUM_K + 128)
#define SMEM_BYTES  (SMEM_FLOATS*4)

// ws layout: [0]=sse, [1..512]=hist, [513..1024]=||w_k||^2
__global__ void vq_init_kernel(const float* __restrict__ W, float* __restrict__ wsf) {
    const int i = threadIdx.x;            // 0..511
    if (i == 0) wsf[0] = 0.0f;
    wsf[1 + i] = 0.0f;
    const float* wr = W + i * DIM;
    float s = 0.0f;
    #pragma unroll
    for (int d = 0; d < DIM; ++d) s += wr[d] * wr[d];
    wsf[513 + i] = s;
}

__global__ __launch_bounds__(256) void vq_main_kernel(const float* __restrict__ X,
                                                      const float* __restrict__ W,
                                                      float* __restrict__ out,
                                                      float* __restrict__ wsf) {
    extern __shared__ float smem[];
    float* w_lds  = smem;                       // [512][68] padded codebook
    float* wn_lds = smem + NUM_K*WROW;          // [512] squared norms
    int*   idx_lds = (int*)(smem + NUM_K*WROW + NUM_K); // [8*16] per-wave indices

    const int tid = threadIdx.x;

    // ---- async-stage codebook into LDS: 8192 x b128 chunks, direct global->LDS ----
    // chunk c covers row k = c>>4, dwords d0 = (c&15)*4 .. +3 (row-contiguous; padding
    // only between rows, so every 16B chunk is contiguous in both global and LDS).
    #pragma unroll
    for (int i = 0; i < 32; ++i) {
        int c  = i * 256 + tid;
        int k  = c >> 4;
        int d0 = (c & 15) << 2;
        unsigned lds_b = (unsigned)(size_t)(&w_lds[k * WROW + d0]); // low 32b = LDS byte addr
        const float* g = W + (c << 2);
        asm volatile("global_load_async_to_lds_b128 %0, %1, off"
                     :: "v"(lds_b), "v"(g) : "memory");
    }
    // stage precomputed ||w||^2 from workspace
    for (int k = tid; k < NUM_K; k += 256) wn_lds[k] = wsf[513 + k];
    asm volatile("s_wait_asynccnt 0x0" ::: "memory");
    __syncthreads();

    const int wave  = tid >> 5;
    const int lane  = tid & 31;
    const int laneN = lane & 15;
    const int half  = lane >> 4;      // 0: lanes 0-15, 1: lanes 16-31
    const int sel   = half * 2;       // A/B K sub-select per 16x16x4 f32 layout

    const int n0  = (blockIdx.x * 8 + wave) * 16;   // 16-row tile
    const int b   = n0 >> 12;
    const int hw0 = n0 & 4095;
    const float* xp = X + (size_t)b*PLANE + hw0 + laneN;

    // ---- load A tile (16 rows x 64 dims): lane holds row laneN, K = 4j+sel, 4j+sel+1 ----
    // X is read exactly once -> non-temporal loads keep it out of the reuse caches.
    v2f a[16];
    float p = 0.0f;
    #pragma unroll
    for (int j = 0; j < 16; ++j) {
        int d0 = 4*j + sel;
        float x0 = __builtin_nontemporal_load(xp + (size_t)d0     * HW);
        float x1 = __builtin_nontemporal_load(xp + (size_t)(d0+1) * HW);
        a[j].x = x0; a[j].y = x1;
        p += x0*x0 + x1*x1;
    }
    float xn = p + __shfl_xor(p, 16, 32);   // lane q holds ||x_{row q%16}||^2
    float xr[8];
    #pragma unroll
    for (int r = 0; r < 8; ++r) xr[r] = __shfl(xn, r + 8*half, 16); // norm for C-slot row

    float bd[8]; int bi[8];
    #pragma unroll
    for (int r = 0; r < 8; ++r) { bd[r] = 3.4e38f; bi[r] = 0; }

    // ---- distance matmul: 32 k-tiles, 16 x V_WMMA_F32_16X16X4_F32 each ----
    for (int kt = 0; kt < 32; ++kt) {
        const int k0 = kt * 16;
        const float* wrow = &w_lds[(k0 + laneN)*WROW + sel];
        // batch B fragments so DS latency overlaps the WMMA chain
        v2f bv[16];
        #pragma unroll
        for (int j = 0; j < 16; ++j) bv[j] = *(const v2f*)(wrow + 4*j);
        v8f c = {0.f,0.f,0.f,0.f,0.f,0.f,0.f,0.f};
        #pragma unroll
        for (int j = 0; j < 16; ++j)
            c = __builtin_amdgcn_wmma_f32_16x16x4_f32(false, a[j], false, bv[j],
                                                      (short)0, c, false, false);
        const float wn = wn_lds[k0 + laneN];
        const int   kk = k0 + laneN;
        #pragma unroll
        for (int r = 0; r < 8; ++r) {
            float d = xr[r] + wn - 2.0f * c[r];
            if (d < bd[r]) { bd[r] = d; bi[r] = kk; }
        }
    }

    // ---- argmin butterfly within each 16-lane half (tie -> smaller index) ----
    #pragma unroll
    for (int m = 1; m <= 8; m <<= 1) {
        #pragma unroll
        for (int r = 0; r < 8; ++r) {
            float od = __shfl_xor(bd[r], m, 32);
            int   oi = __shfl_xor(bi[r], m, 32);
            if (od < bd[r] || (od == bd[r] && oi < bi[r])) { bd[r] = od; bi[r] = oi; }
        }
    }
    if (lane == 0) {
        #pragma unroll
        for (int r = 0; r < 8; ++r) idx_lds[wave*16 + r] = bi[r];
    }
    if (lane == 16) {
        #pragma unroll
        for (int r = 0; r < 8; ++r) idx_lds[wave*16 + 8 + r] = bi[r];
    }
    __syncthreads();

    const int idxrow = idx_lds[wave*16 + laneN];  // code index for row laneN of tile

    // ---- histogram (one add per row) ----
    if (lane < 16) atomicAdd(&wsf[1 + idxrow], 1.0f);

    // ---- quantized NCHW output + SSE straight from A registers ----
    // lane covers (row laneN, channels 4j+sel, 4j+sel+1): same mapping as A tile,
    // so no X reload; stores stay 64B-coalesced per 16-lane half. Write-once ->
    // non-temporal stores.
    float sse = 0.0f;
    float* outp = out + OUT_OFF;
    const size_t obase = (size_t)b*PLANE + hw0 + laneN;
    #pragma unroll
    for (int j = 0; j < 16; ++j) {
        int c0 = 4*j + sel;
        v2f q = *(const v2f*)&w_lds[idxrow*WROW + c0];   // ds_load_b64
        __builtin_nontemporal_store(q.x, outp + obase + (size_t)c0    *HW);
        __builtin_nontemporal_store(q.y, outp + obase + (size_t)(c0+1)*HW);
        float d0 = q.x - a[j].x;
        float d1 = q.y - a[j].y;
        sse += d0*d0 + d1*d1;
    }
    #pragma unroll
    for (int m = 1; m < 32; m <<= 1) sse += __shfl_xor(sse, m, 32);
    if (lane == 0) atomicAdd(&wsf[0], sse);

    // ---- one-hot encodings: 268MB streaming write, non-temporal v2f stores ----
    float* enc = out + ENC_OFF;
    #pragma unroll
    for (int mrow = 0; mrow < 16; ++mrow) {
        int im = idx_lds[wave*16 + mrow];
        float* eb = enc + (size_t)(n0 + mrow) * NUM_K;
        #pragma unroll
        for (int t = 0; t < 8; ++t) {
            int col = lane*2 + 64*t;
            v2f v;
            v.x = (col     == im) ? 1.0f : 0.0f;
            v.y = (col + 1 == im) ? 1.0f : 0.0f;
            __builtin_nontemporal_store(v, (v2f*)(eb + col));
        }
    }
}

__global__ void vq_final_kernel(const float* __restrict__ wsf, float* __restrict__ out) {
    __shared__ float red[256];
    int t = threadIdx.x;
    float local = 0.0f;
    for (int k = t; k < NUM_K; k += 256) {
        float pk = wsf[1 + k] * (1.0f / 131072.0f);
        local += pk * logf(pk + 1e-10f);
    }
    red[t] = local;
    __syncthreads();
    for (int s = 128; s > 0; s >>= 1) {
        if (t < s) red[t] += red[t + s];
        __syncthreads();
    }
    if (t == 0) {
        out[PERP_OFF] = expf(-red[0]);
        out[0] = 1.25f * wsf[0] / NELEM;   // q_loss + 0.25*e_loss, both == MSE
    }
}

extern "C" void kernel_launch(void* const* d_in, const int* in_sizes, int n_in,
                              void* d_out, int out_size, void* d_ws, size_t ws_size,
                              hipStream_t stream) {
    const float* X = (const float*)d_in[0];   // [32,64,64,64] NCHW
    const float* W = (const float*)d_in[1];   // [512,64]
    float* out = (float*)d_out;
    float* wsf = (float*)d_ws;                // [0]=sse, [1..512]=hist, [513..1024]=wnorm

    (void)in_sizes; (void)n_in; (void)out_size; (void)ws_size;

    // Opt in to >64KB dynamic LDS (idempotent; not a stream op, capture-safe).
    hipFuncSetAttribute(reinterpret_cast<const void*>(vq_main_kernel),
                        hipFuncAttributeMaxDynamicSharedMemorySize, SMEM_BYTES);

    vq_init_kernel<<<1, 512, 0, stream>>>(W, wsf);
    vq_main_kernel<<<1024, 256, SMEM_BYTES, stream>>>(X, W, out, wsf);
    vq_final_kernel<<<1, 256, 0, stream>>>(wsf, out);
}